// PHMEmbedding_84456236909170
// MI455X (gfx1250) — compile-verified
//
#include <hip/hip_runtime.h>

// PHM embedding: out[t, q*256+j] = sum_k a[k,p,q] * s[k,i,j],
//   p = idx[t] / 12565, i = idx[t] % 12565, q in [0,4), j in [0,256).
// One wave32 per token; 16x V_WMMA_F32_16X16X4_F32 per token (K=4 == k-sum).
// Memory-bound (~128 MiB traffic, ~5.5us at 23.3 TB/s); WMMA is roofline-free.
//
// Fragment mapping (wave32, ISA 7.12.2):
//   A 16x4 f32 (2 VGPR): lanes 0-15 hold K={0,1}, lanes 16-31 K={2,3};
//     A[m,k] = a[k, p, m&3]  (q replicated every 4 M-rows)
//   B 4x16 f32 (2 VGPR): same lane-half K split; B[k,n] = s[k, i, j0+n]
//   D 16x16 f32 (8 VGPR): C[v] holds M = v + (lane<16 ? 0 : 8);
//     since q = M&3, C[0..3] = out[q=0..3, j0+n] on EVERY lane (duplicated
//     across halves). Lanes 0-15 store q=0,1; lanes 16-31 store q=2,3 via
//     two v_cndmask selects (constant vector indices -> no extract chains).

typedef float v2f __attribute__((ext_vector_type(2)));
typedef float v8f __attribute__((ext_vector_type(8)));

#define NROWS   12565          // VOCAB_PAD / PHM
#define ECOLS   256            // EMB_PAD / PHM
#define EMB     1024
#define NE      (NROWS * ECOLS)
#define WAVES_PER_BLOCK 8

__global__ __launch_bounds__(256) void phm_embed_wmma_kernel(
    const int*   __restrict__ idx,   // [ntok]
    const float* __restrict__ a,     // [4,4,4]  (k,p,q)
    const float* __restrict__ s,     // [4,12565,256] (k,i,j)
    float*       __restrict__ out,   // [ntok, 1024]
    int ntok)
{
    const int lane  = threadIdx.x & 31;
    const int wave  = threadIdx.x >> 5;
    const int token = blockIdx.x * WAVES_PER_BLOCK + wave;
    if (token >= ntok) return;                 // wave-uniform: EXEC stays all-1s

    const int t = idx[token];
    const int p = t / NROWS;                   // 0..3
    const int i = t - p * NROWS;               // 0..12564

    const int  n     = lane & 15;              // N column within 16-wide tile
    const bool hi    = lane >= 16;
    const int  kb    = hi ? 2 : 0;             // K-half held by this lane half
    const int  qbase = hi ? 2 : 0;             // q rows this lane stores
    const int  q     = n & 3;                  // A rows replicate q mod 4

    // A fragment: A[m,k] = a[k, p, m&3]
    v2f A;
    A.x = a[(kb + 0) * 16 + p * 4 + q];
    A.y = a[(kb + 1) * 16 + p * 4 + q];

    // Hoisted per-lane base pointers; per-tile accesses use immediate offsets.
    const float* pl0 = s + (size_t)(kb + 0) * NE + (size_t)i * ECOLS + n;
    const float* pl1 = s + (size_t)(kb + 1) * NE + (size_t)i * ECOLS + n;
    float*       pst = out + (size_t)token * EMB + qbase * ECOLS + n;

#pragma unroll
    for (int tile = 0; tile < 16; ++tile) {
        const int j0 = tile * 16;

        v2f B;
        B.x = pl0[j0];
        B.y = pl1[j0];

        v8f C = {};
        // (neg_a, A, neg_b, B, c_mod, C, reuse_a, reuse_b)
        C = __builtin_amdgcn_wmma_f32_16x16x4_f32(
                false, A, false, B, (short)0, C, false, false);

        // Constant-index extracts + lane-half select (2x v_cndmask):
        const float va = hi ? C[2] : C[0];     // q = qbase + 0
        const float vb = hi ? C[3] : C[1];     // q = qbase + 1
        pst[j0]         = va;
        pst[ECOLS + j0] = vb;
    }
}

extern "C" void kernel_launch(void* const* d_in, const int* in_sizes, int n_in,
                              void* d_out, int out_size, void* d_ws, size_t ws_size,
                              hipStream_t stream) {
    const int*   idx = (const int*)  d_in[0];   // [8*2048] int32
    const float* a   = (const float*)d_in[1];   // [4,4,4] f32
    const float* s   = (const float*)d_in[2];   // [4,12565,256] f32
    float*       out = (float*)      d_out;     // [8*2048, 1024] f32

    const int ntok   = in_sizes[0];                              // 16384
    const int blocks = (ntok + WAVES_PER_BLOCK - 1) / WAVES_PER_BLOCK;

    phm_embed_wmma_kernel<<<blocks, 256, 0, stream>>>(idx, a, s, out, ntok);
}